// FFTToroidalAttention_37641093382538
// MI455X (gfx1250) — compile-verified
//
#include <hip/hip_runtime.h>
#include <hip/hip_bf16.h>

// ---------------------------------------------------------------------------
// FFT circulant attention for MI455X (gfx1250), fp32 end-to-end.
//   Projections: V_WMMA_F32_16X16X4_F32, global->LDS staging via async
//   load-to-LDS (b128/b32) when the toolchain exposes the builtins.
//   Sequence FFTs (N=8192): single-workgroup radix-2 DIT in 64KB LDS.
// ---------------------------------------------------------------------------

#define SEQ_N   8192
#define DMODEL  1024
#define NBATCH  2
#define NHEADS  16
#define DK      64

typedef float v2f  __attribute__((ext_vector_type(2)));
typedef float v8f  __attribute__((ext_vector_type(8)));
typedef int   i32x4 __attribute__((ext_vector_type(4)));

#ifndef __has_builtin
#define __has_builtin(x) 0
#endif
#if __has_builtin(__builtin_amdgcn_global_load_async_to_lds_b128) && \
    __has_builtin(__builtin_amdgcn_global_load_async_to_lds_b32)
#define USE_ASYNC_LDS 1
#endif

#define AS1 __attribute__((address_space(1)))
#define AS3 __attribute__((address_space(3)))

// 16-byte global -> LDS copy (async when available)
__device__ __forceinline__ void cp16(const float* g, float* l) {
#ifdef USE_ASYNC_LDS
    __builtin_amdgcn_global_load_async_to_lds_b128((AS1 i32x4*)g, (AS3 i32x4*)l, 0, 0);
#else
    *(float4*)l = *(const float4*)g;
#endif
}
// 4-byte global -> LDS copy (async when available)
__device__ __forceinline__ void cp4(const float* g, float* l) {
#ifdef USE_ASYNC_LDS
    __builtin_amdgcn_global_load_async_to_lds_b32((AS1 int*)g, (AS3 int*)l, 0, 0);
#else
    *l = *g;
#endif
}
__device__ __forceinline__ void async_fence() {
#ifdef USE_ASYNC_LDS
    asm volatile("s_wait_asynccnt 0" ::: "memory");
#endif
}

// ---------------------------------------------------------------------------
// GEMM: out = A(16384x1024) * W^T(1024x1024) + bias
//   MODE 0: A row-major (x), output stored transposed: out[(b*E+e)*N + n]
//   MODE 1: A stored transposed (Ot[(b*E+k)*N + n]), output row-major
// Block: 128 threads (4 waves). Tile 128(M) x 64(E), K in 16-slabs.
// Each wave: 32 rows x 64 cols -> 8 WMMA per k-step (B frag reused 2x).
// LDS tiles are [row][k] with row stride 20 floats: global-contiguous k maps
// to LDS-contiguous bytes (async b128 friendly), fragment reads are b64 and
// bank-conflict free (gcd(20,64)=4, 16 lanes distinct, halves disjoint mod 4).
// ---------------------------------------------------------------------------
template <int MODE>
__global__ __launch_bounds__(128)
void gemm_wmma_kernel(const float* __restrict__ A,
                      const float* __restrict__ W,
                      const float* __restrict__ bias,
                      float* __restrict__ out)
{
    const int bm   = blockIdx.x * 128;    // global row start (m over B*N)
    const int bn   = blockIdx.y * 64;     // output col start (e)
    const int tid  = threadIdx.x;         // 0..127
    const int wave = tid >> 5;            // 0..3
    const int lane = tid & 31;

    __shared__ __align__(16) float As[128][20];   // [m][k], 10.0 KB
    __shared__ __align__(16) float Bs[64][20];    // [e][k],  5.0 KB
    __shared__ float Cs[128][69];                 // result bounce, 34.5 KB

    v8f acc[2][4] = {};                   // 2 M-tiles x 4 E-tiles of 16x16

    const int bb = bm >> 13;              // batch of this row block
    const int n0 = bm & (SEQ_N - 1);      // sequence offset within batch

    for (int k0 = 0; k0 < DMODEL; k0 += 16) {
        // ---- stage A slab: 128(m) x 16(k) ----
        if (MODE == 0) {
            #pragma unroll
            for (int i = 0; i < 4; ++i) {
                const int q = tid + (i << 7);          // [0,512) float4s
                const int m = q >> 2, c = q & 3;
                cp16(&A[(size_t)(bm + m) * DMODEL + k0 + 4 * c], &As[m][4 * c]);
            }
        } else {
            #pragma unroll
            for (int k = 0; k < 16; ++k)               // lanes sweep n: coalesced
                cp4(&A[((size_t)bb * DMODEL + k0 + k) * SEQ_N + n0 + tid], &As[tid][k]);
        }
        // ---- stage W slab: Bs[e][k] = W[bn+e][k0+k] ----
        #pragma unroll
        for (int i = 0; i < 2; ++i) {
            const int q = tid + (i << 7);              // [0,256) float4s
            const int e = q >> 2, c = q & 3;
            cp16(&W[(size_t)(bn + e) * DMODEL + k0 + 4 * c], &Bs[e][4 * c]);
        }
        async_fence();
        __syncthreads();

        // ---- compute: 4 k-steps of 4, 8 WMMA each ----
        #pragma unroll
        for (int kk = 0; kk < 16; kk += 4) {
            // frag K index: lane half selects K pair (r + 2*(lane>=16))
            const int kb   = kk + ((lane >> 4) << 1);
            const int lrow = (wave << 5) + (lane & 15);
            const v2f a0 = *(const v2f*)&As[lrow][kb];
            const v2f a1 = *(const v2f*)&As[lrow + 16][kb];
            #pragma unroll
            for (int c = 0; c < 4; ++c) {
                const v2f bf = *(const v2f*)&Bs[(c << 4) + (lane & 15)][kb];
                acc[0][c] = __builtin_amdgcn_wmma_f32_16x16x4_f32(
                    false, a0, false, bf, (short)0, acc[0][c], false, false);
                acc[1][c] = __builtin_amdgcn_wmma_f32_16x16x4_f32(
                    false, a1, false, bf, (short)0, acc[1][c], false, false);
            }
        }
        __syncthreads();
    }

    // ---- epilogue: C frag (VGPR r, lane l -> M=r+8*(l>=16), N=l%16) -> LDS ----
    #pragma unroll
    for (int mt = 0; mt < 2; ++mt) {
        #pragma unroll
        for (int c = 0; c < 4; ++c) {
            const int e  = (c << 4) + (lane & 15);
            const float bv = bias ? bias[bn + e] : 0.0f;
            #pragma unroll
            for (int r = 0; r < 8; ++r) {
                const int m = (wave << 5) + (mt << 4) + r + ((lane >> 4) << 3);
                Cs[m][e] = acc[mt][c][r] + bv;
            }
        }
    }
    __syncthreads();

    // ---- coalesced b128 stores from LDS ----
    if (MODE == 0) {
        // out[(b*E + e)*N + n]: 128 consecutive n per e-row
        for (int q = tid; q < 128 * 16; q += 128) {    // 2048 float4s
            const int e = q >> 5, c4 = q & 31;
            float4 v;
            v.x = Cs[4 * c4 + 0][e];
            v.y = Cs[4 * c4 + 1][e];
            v.z = Cs[4 * c4 + 2][e];
            v.w = Cs[4 * c4 + 3][e];
            *(float4*)&out[((size_t)bb * DMODEL + bn + e) * SEQ_N + n0 + 4 * c4] = v;
        }
    } else {
        // out[m*E + e]: 64 consecutive e per m-row
        for (int q = tid; q < 128 * 16; q += 128) {
            const int m = q >> 4, c4 = q & 15;
            float4 v;
            v.x = Cs[m][4 * c4 + 0];
            v.y = Cs[m][4 * c4 + 1];
            v.z = Cs[m][4 * c4 + 2];
            v.w = Cs[m][4 * c4 + 3];
            *(float4*)&out[(size_t)(bm + m) * DMODEL + bn + 4 * c4] = v;
        }
    }
}

// ---------------------------------------------------------------------------
// 8192-point complex radix-2 DIT FFT, one sequence per workgroup, LDS-resident.
//   inR != nullptr : real input (imag = 0), else complex inC
//   outR != nullptr: write real part * scale, else complex outC * scale
//   mulF != nullptr: frequency-domain multiply (row = seq>>6) fused at load
//   dirSign: -1 forward (numpy fft), +1 inverse (scale should carry 1/N)
// ---------------------------------------------------------------------------
__device__ __forceinline__ float2 cmulf(float2 a, float2 b) {
    return make_float2(a.x * b.x - a.y * b.y, a.x * b.y + a.y * b.x);
}

__global__ __launch_bounds__(256)
void fft8192_kernel(const float*  __restrict__ inR,
                    const float2* __restrict__ inC,
                    float*        __restrict__ outR,
                    float2*       __restrict__ outC,
                    const float2* __restrict__ mulF,
                    float dirSign, float scale)
{
    __shared__ float2 lds[SEQ_N];                 // 64 KB
    const int    seq  = blockIdx.x;
    const int    tid  = threadIdx.x;
    const size_t base = (size_t)seq * SEQ_N;

    // Bit-reversed load (13-bit), optional freq-domain multiply
    for (int i = tid; i < SEQ_N; i += 256) {
        const int j = (int)(__brev((unsigned)i) >> 19);
        float2 v = inR ? make_float2(inR[base + j], 0.0f) : inC[base + j];
        if (mulF) v = cmulf(v, mulF[(size_t)(seq >> 6) * SEQ_N + j]);
        lds[i] = v;
    }
    __syncthreads();

    const float twoPi = 6.283185307179586f;
    for (int len = 2; len <= SEQ_N; len <<= 1) {
        const int   half = len >> 1;
        const int   mask = half - 1;
        const float ang0 = dirSign * twoPi / (float)len;
        for (int j = tid; j < (SEQ_N >> 1); j += 256) {
            const int pos = j & mask;
            const int i0  = ((j & ~mask) << 1) | pos;
            const int i1  = i0 + half;
            float s, c;
            __sincosf(ang0 * (float)pos, &s, &c);
            const float2 w = make_float2(c, s);
            const float2 t = cmulf(w, lds[i1]);
            const float2 u = lds[i0];
            lds[i1] = make_float2(u.x - t.x, u.y - t.y);
            lds[i0] = make_float2(u.x + t.x, u.y + t.y);
        }
        __syncthreads();
    }

    if (outR) {
        for (int i = tid; i < SEQ_N; i += 256)
            outR[base + i] = lds[i].x * scale;
    } else {
        for (int i = tid; i < SEQ_N; i += 256) {
            const float2 v = lds[i];
            outC[base + i] = make_float2(v.x * scale, v.y * scale);
        }
    }
}

// ---------------------------------------------------------------------------
// scores_fft[b][h][f] = sum_d Qf[b][h*64+d][f] * conj(Kf[b][h*64+d][f])
// ---------------------------------------------------------------------------
__global__ __launch_bounds__(256)
void score_freq_kernel(const float2* __restrict__ Qf,
                       const float2* __restrict__ Kf,
                       float2* __restrict__ Sf)
{
    const int idx = blockIdx.x * 256 + threadIdx.x;   // over B*H*N
    const int f   = idx & (SEQ_N - 1);
    const int bh  = idx >> 13;
    const int b   = bh >> 4;
    const int h   = bh & 15;
    const size_t base = ((size_t)b * DMODEL + (size_t)h * DK) * SEQ_N + f;

    float ax = 0.0f, ay = 0.0f;
    #pragma unroll 4
    for (int d = 0; d < DK; ++d) {
        const float2 q = Qf[base + (size_t)d * SEQ_N];
        const float2 k = Kf[base + (size_t)d * SEQ_N];
        ax += q.x * k.x + q.y * k.y;      // q * conj(k)
        ay += q.y * k.x - q.x * k.y;
    }
    Sf[(size_t)bh * SEQ_N + f] = make_float2(ax, ay);
}

// ---------------------------------------------------------------------------
// Row softmax over 8192 elements, one block per (b,h) row.
// ---------------------------------------------------------------------------
__global__ __launch_bounds__(256)
void softmax8192_kernel(const float* __restrict__ in, float* __restrict__ out)
{
    __shared__ float red[256];
    const int row = blockIdx.x;
    const int tid = threadIdx.x;
    const float* p = in  + (size_t)row * SEQ_N;
    float*       q = out + (size_t)row * SEQ_N;

    float m = -3.4e38f;
    for (int i = tid; i < SEQ_N; i += 256) m = fmaxf(m, p[i]);
    red[tid] = m; __syncthreads();
    for (int s = 128; s > 0; s >>= 1) {
        if (tid < s) red[tid] = fmaxf(red[tid], red[tid + s]);
        __syncthreads();
    }
    m = red[0]; __syncthreads();

    float sum = 0.0f;
    for (int i = tid; i < SEQ_N; i += 256) sum += __expf(p[i] - m);
    red[tid] = sum; __syncthreads();
    for (int s = 128; s > 0; s >>= 1) {
        if (tid < s) red[tid] += red[tid + s];
        __syncthreads();
    }
    const float inv = 1.0f / red[0];
    for (int i = tid; i < SEQ_N; i += 256) q[i] = __expf(p[i] - m) * inv;
}

// ---------------------------------------------------------------------------
// Host-side launch sequence
// ---------------------------------------------------------------------------
extern "C" void kernel_launch(void* const* d_in, const int* in_sizes, int n_in,
                              void* d_out, int out_size, void* d_ws, size_t ws_size,
                              hipStream_t stream)
{
    (void)in_sizes; (void)n_in; (void)out_size; (void)ws_size;

    const float* x  = (const float*)d_in[0];
    const float* Wq = (const float*)d_in[1];
    const float* bq = (const float*)d_in[2];
    const float* Wk = (const float*)d_in[3];
    const float* bk = (const float*)d_in[4];
    const float* Wv = (const float*)d_in[5];
    const float* bv = (const float*)d_in[6];
    const float* Wo = (const float*)d_in[7];
    const float* bo = (const float*)d_in[8];

    char* ws = (char*)d_ws;
    const size_t R = (size_t)NBATCH * DMODEL * SEQ_N * sizeof(float);   // 64 MB

    float*  Qt     = (float* )(ws + 0 * R);   // (b, e, n) real
    float*  Kt     = (float* )(ws + 1 * R);
    float*  Vt     = (float* )(ws + 2 * R);
    float2* Qf     = (float2*)(ws + 3 * R);   // (b, e, f) complex
    float2* Kf     = (float2*)(ws + 5 * R);
    float2* Vf     = (float2*)(ws + 7 * R);
    float2* Sf     = (float2*)(ws + 9 * R);                       // (bh, f)
    float*  scores = (float* )(ws + 9 * R + (size_t)(3u << 20));  // (bh, n)
    float*  attn   = (float* )(ws + 9 * R + (size_t)(4u << 20));
    float2* attnF  = (float2*)(ws + 9 * R + (size_t)(5u << 20));
    float*  Ot     = Qt;   // Qt is dead after Qf is produced

    const dim3 gGemm((NBATCH * SEQ_N) / 128, DMODEL / 64), bGemm(128);
    const int  nSeqQKV = NBATCH * DMODEL;   // 2048 sequences
    const int  nSeqBH  = NBATCH * NHEADS;   // 32 sequences

    // 1) Projections (WMMA), stored sequence-contiguous.
    gemm_wmma_kernel<0><<<gGemm, bGemm, 0, stream>>>(x, Wq, bq, Qt);
    gemm_wmma_kernel<0><<<gGemm, bGemm, 0, stream>>>(x, Wk, bk, Kt);
    gemm_wmma_kernel<0><<<gGemm, bGemm, 0, stream>>>(x, Wv, bv, Vt);

    // 2) Forward FFTs along sequence axis.
    fft8192_kernel<<<nSeqQKV, 256, 0, stream>>>(Qt, nullptr, nullptr, Qf, nullptr, -1.0f, 1.0f);
    fft8192_kernel<<<nSeqQKV, 256, 0, stream>>>(Kt, nullptr, nullptr, Kf, nullptr, -1.0f, 1.0f);
    fft8192_kernel<<<nSeqQKV, 256, 0, stream>>>(Vt, nullptr, nullptr, Vf, nullptr, -1.0f, 1.0f);

    // 3) scores_fft = sum_d Qf * conj(Kf); inverse FFT with 1/N and 1/sqrt(dk).
    score_freq_kernel<<<(NBATCH * NHEADS * SEQ_N) / 256, 256, 0, stream>>>(Qf, Kf, Sf);
    fft8192_kernel<<<nSeqBH, 256, 0, stream>>>(nullptr, Sf, scores, nullptr, nullptr,
                                               +1.0f, (1.0f / SEQ_N) * 0.125f);

    // 4) softmax -> attn; forward FFT of attn.
    softmax8192_kernel<<<nSeqBH, 256, 0, stream>>>(scores, attn);
    fft8192_kernel<<<nSeqBH, 256, 0, stream>>>(attn, nullptr, nullptr, attnF, nullptr, -1.0f, 1.0f);

    // 5) O = irfft(attn_fft * Vf): multiply fused into inverse-FFT load.
    fft8192_kernel<<<nSeqQKV, 256, 0, stream>>>(nullptr, Vf, Ot, nullptr, attnF,
                                                +1.0f, 1.0f / SEQ_N);

    // 6) Output projection (WMMA) from transposed O, row-major result + bias.
    gemm_wmma_kernel<1><<<gGemm, bGemm, 0, stream>>>(Ot, Wo, bo, (float*)d_out);
}